// GlobalKMaxPool2d_1752346657517
// MI455X (gfx1250) — compile-verified
//
#include <hip/hip_runtime.h>
#include <cstdint>

// Problem geometry (fixed by the reference): x is (64,256,56,56) f32, K=8.
#define HW      3136     // 56*56 floats per row
#define TILE_F  128      // floats per async tile: 32 lanes * 4 floats (b128/lane)
#define TILE_B  512      // bytes per tile
#define NTILES  24       // 24*128 = 3072 floats via async pipeline
#define TAIL0   3072     // remaining 64 floats loaded directly
#define PF      8        // async pipeline depth (LDS buffers per wave)
#define WAVES   8        // waves (rows) per 256-thread block
#define KSEL    8

// Branchless sorted-descending top-8 insert: 8 dependent (max,min) pairs,
// the max/min in each pair are independent -> VOPD dual-issue pairable.
__device__ __forceinline__ void insert8(float (&tk)[KSEL], float x) {
#pragma unroll
    for (int j = 0; j < KSEL; ++j) {
        float old = tk[j];
        tk[j] = fmaxf(old, x);
        x     = fminf(old, x);
    }
}

// CDNA5 async global->LDS load (ASYNCcnt-tracked DMA path).
// GVS addressing: mem = SGPR64 base + VGPR32 byte offset; VDST = LDS byte addr.
__device__ __forceinline__ void async_tile(unsigned lds_byte_addr,
                                           unsigned long long gbase,
                                           unsigned voff_bytes) {
    asm volatile("global_load_async_to_lds_b128 %0, %1, %2"
                 :: "v"(lds_byte_addr), "v"(voff_bytes), "s"(gbase)
                 : "memory");
}

// Wait until at most N async ops remain outstanding (constant immediate).
__device__ __forceinline__ void wait_async_le(int w) {
    switch (w) {
        case 7:  asm volatile("s_wait_asynccnt 0x7" ::: "memory"); break;
        case 6:  asm volatile("s_wait_asynccnt 0x6" ::: "memory"); break;
        case 5:  asm volatile("s_wait_asynccnt 0x5" ::: "memory"); break;
        case 4:  asm volatile("s_wait_asynccnt 0x4" ::: "memory"); break;
        case 3:  asm volatile("s_wait_asynccnt 0x3" ::: "memory"); break;
        case 2:  asm volatile("s_wait_asynccnt 0x2" ::: "memory"); break;
        case 1:  asm volatile("s_wait_asynccnt 0x1" ::: "memory"); break;
        default: asm volatile("s_wait_asynccnt 0x0" ::: "memory"); break;
    }
}

__global__ __launch_bounds__(256, 1)
void topk8_rowsum(const float* __restrict__ x, float* __restrict__ out, int nrows) {
    __shared__ __align__(16) float tiles[WAVES][PF][TILE_F];

    const int lane = threadIdx.x & 31;   // wave32
    const int wave = threadIdx.x >> 5;
    const int row  = blockIdx.x * WAVES + wave;
    if (row >= nrows) return;            // wave-uniform, no barriers used

    const float* rowp = x + (size_t)row * HW;
    const unsigned long long gbase = (unsigned long long)(uintptr_t)rowp; // wave-uniform

    // Per-lane LDS byte addresses of the PF staging buffers
    // (low 32 bits of a generic pointer to __shared__ = LDS offset).
    const unsigned lds0 = (unsigned)(uintptr_t)(&tiles[wave][0][lane * 4]);

    float tk[KSEL];
#pragma unroll
    for (int j = 0; j < KSEL; ++j) tk[j] = -__builtin_inff();

    // Prologue: fill the async pipeline (PF tiles in flight).
#pragma unroll
    for (int p = 0; p < PF; ++p)
        async_tile(lds0 + (unsigned)(p * TILE_B), gbase,
                   (unsigned)(p * TILE_B + lane * 16));

    // Fully unrolled: every wait immediate folds to a constant, no loop SALU,
    // and the backend can pipeline ds_load -> insert chains across tiles.
#pragma unroll
    for (int t = 0; t < NTILES; ++t) {
        // Async loads complete in order; tile t is done once outstanding
        // drops to (#issued - (t+1)).
        const int rem = NTILES - t - 1;
        wait_async_le(rem < (PF - 1) ? rem : (PF - 1));

        const float4 v =
            *reinterpret_cast<const float4*>(&tiles[wave][t & (PF - 1)][lane * 4]);
        insert8(tk, v.x); insert8(tk, v.y); insert8(tk, v.z); insert8(tk, v.w);

        // Refill only after the ds read was consumed (no WAR on the buffer).
        const int nt = t + PF;
        if (nt < NTILES)
            async_tile(lds0 + (unsigned)((nt & (PF - 1)) * TILE_B), gbase,
                       (unsigned)(nt * TILE_B + lane * 16));
    }

    // Tail: 64 floats, direct nontemporal (single-pass streaming) loads.
    {
        const float* tp = rowp + TAIL0 + lane * 2;
        float a = __builtin_nontemporal_load(tp);
        float b = __builtin_nontemporal_load(tp + 1);
        insert8(tk, a); insert8(tk, b);
    }

    // Wave32 merge: the global top-8 are among the 32 sorted per-lane heads.
    // 8 rounds: argmax over heads (5 shfl_xor steps on (val,lane)), pop winner.
    float sum = 0.0f;
#pragma unroll
    for (int r = 0; r < KSEL; ++r) {
        float m  = tk[0];
        int   mi = lane;
#pragma unroll
        for (int off = 16; off > 0; off >>= 1) {
            float om = __shfl_xor(m, off, 32);
            int   oi = __shfl_xor(mi, off, 32);
            if (om > m || (om == m && oi < mi)) { m = om; mi = oi; }
        }
        sum += m;
        if (lane == mi) {
#pragma unroll
            for (int j = 0; j < KSEL - 1; ++j) tk[j] = tk[j + 1];
            tk[KSEL - 1] = -__builtin_inff();
        }
    }

    if (lane == 0) out[row] = sum;
}

extern "C" void kernel_launch(void* const* d_in, const int* in_sizes, int n_in,
                              void* d_out, int out_size, void* d_ws, size_t ws_size,
                              hipStream_t stream) {
    (void)n_in; (void)d_ws; (void)ws_size;
    const float* x  = (const float*)d_in[0];
    float* out      = (float*)d_out;
    const int total = in_sizes[0];        // 64*256*56*56
    const int nrows = total / HW;         // 16384 (== out_size)
    (void)out_size;

    dim3 block(256);
    dim3 grid((nrows + WAVES - 1) / WAVES);
    hipLaunchKernelGGL(topk8_rowsum, grid, block, 0, stream, x, out, nrows);
}